// AGCN_17497696764343
// MI455X (gfx1250) — compile-verified
//
#include <hip/hip_runtime.h>
#include <hip/hip_bf16.h>

// AGCN: B=32, N=2000, DIM_IN=DIM_OUT=64, EMBED=16, CHEB_K=3
//   S  = softmax(relu(nv1@nv2), axis=1)            [N,N]   (kernel 1)
//   Y1 = S @ x      (per batch)                    [B,N,C] (kernel 2, WMMA f32)
//   Y2 = 2*S @ Y1 - x                              [B,N,C] (kernel 3, same kernel)
//   out[b,n,:] = [x|Y1|Y2][b,n,:] @ W_n(192x64) + bias_n   (kernel 4, WMMA f32)
//   with W_n = sum_d emb[n,d]*pool[d] built in LDS per node.
//
// WMMA f32 16x16x4 fragments (ISA 7.12.2):
//   A 16x4 : lane<16 -> M=lane,   K={kb,kb+1} with kb=0 ; lane>=16 -> kb=2
//   B 4x16 : lane<16 -> N=lane,   K={kb,kb+1} with kb=0 ; lane>=16 -> kb=2
//   C 16x16: VGPR j, lane<16 -> M=j, lane>=16 -> M=j+8 ; N=lane&15
// B fragments are read as single 8B-aligned ds_load_b64 from K-contiguous
// (transposed) LDS tiles so the pair lands directly in an aligned VGPR pair.

typedef __attribute__((ext_vector_type(2))) float v2f;
typedef __attribute__((ext_vector_type(8))) float v8f;

#define Nn 2000
#define Cc 64
#define Bb 32
#define Ee 16
#define KI 192           // CHEB_K * DIM_IN
#define XT_STRIDE 66     // transposed chunk: ldsT[col*66 + k]; even -> b64 aligned
#define WT_STRIDE 194    // transposed weights: lwT[o*194 + k]; even -> b64 aligned

// ---------------------------------------------------------------------------
// Kernel 1: adjacency  S = softmax(relu(nv1 @ nv2), axis=1).  One block / row.
// ---------------------------------------------------------------------------
__global__ __launch_bounds__(256) void agcn_adj(const float* __restrict__ nv1,
                                                const float* __restrict__ nv2,
                                                float* __restrict__ S) {
  __shared__ float row[Nn];
  __shared__ float red[256];
  __shared__ float e[Ee];
  const int n = blockIdx.x;
  const int t = threadIdx.x;

  if (t < Ee) e[t] = nv1[n * Ee + t];
  __syncthreads();

  float lmax = 0.0f;  // relu output >= 0
  for (int j = t; j < Nn; j += 256) {
    float z = 0.0f;
#pragma unroll
    for (int d = 0; d < Ee; ++d) z += e[d] * nv2[d * Nn + j];
    z = fmaxf(z, 0.0f);
    row[j] = z;
    lmax = fmaxf(lmax, z);
  }
  red[t] = lmax;
  __syncthreads();
  for (int off = 128; off > 0; off >>= 1) {
    if (t < off) red[t] = fmaxf(red[t], red[t + off]);
    __syncthreads();
  }
  const float m = red[0];
  __syncthreads();

  float lsum = 0.0f;
  for (int j = t; j < Nn; j += 256) {
    float p = expf(row[j] - m);
    row[j] = p;
    lsum += p;
  }
  red[t] = lsum;
  __syncthreads();
  for (int off = 128; off > 0; off >>= 1) {
    if (t < off) red[t] += red[t + off];
    __syncthreads();
  }
  const float inv = 1.0f / red[0];
  for (int j = t; j < Nn; j += 256) S[n * Nn + j] = row[j] * inv;
}

// ---------------------------------------------------------------------------
// Kernel 2/3 helpers: chunk prefetch into registers, transposed store to LDS.
// Chunk = rows [k0, k0+kc) x 64 cols of W[b]; thread t handles row r = t>>2,
// float4 quads q, q+4, q+8, q+12 (q = t&3).  Transposed store -> conflict-free.
// ---------------------------------------------------------------------------
__device__ __forceinline__ void spmm_load_regs(const float* __restrict__ Wb,
                                               int k0, int kc, float4 pre[4]) {
  const int r = threadIdx.x >> 2;
  const int q = threadIdx.x & 3;
  if (r < kc) {
    const float4* src = (const float4*)(Wb + (k0 + r) * Cc);
#pragma unroll
    for (int it = 0; it < 4; ++it) pre[it] = src[q + it * 4];
  } else {
#pragma unroll
    for (int it = 0; it < 4; ++it) pre[it] = make_float4(0.f, 0.f, 0.f, 0.f);
  }
}

__device__ __forceinline__ void spmm_store_lds(float* __restrict__ buf,
                                               const float4 pre[4]) {
  const int r = threadIdx.x >> 2;
  const int q = threadIdx.x & 3;
#pragma unroll
  for (int it = 0; it < 4; ++it) {
    const int c0 = 4 * (q + it * 4);
    buf[(c0 + 0) * XT_STRIDE + r] = pre[it].x;
    buf[(c0 + 1) * XT_STRIDE + r] = pre[it].y;
    buf[(c0 + 2) * XT_STRIDE + r] = pre[it].z;
    buf[(c0 + 3) * XT_STRIDE + r] = pre[it].w;
  }
}

// ---------------------------------------------------------------------------
// Kernel 2/3: Y[b] = alpha * (S @ W[b]) + beta * V[b]        (WMMA f32 16x16x4)
// Block: 256 thr = 8 waves; wave: 16 rows x 64 cols; block: 128 rows.
// Grid: (ceil(N/128), B).  Double-buffered transposed 64x64 chunk in LDS.
// ---------------------------------------------------------------------------
__global__ __launch_bounds__(256) void agcn_spmm(const float* __restrict__ S,
                                                 const float* __restrict__ W,
                                                 float* __restrict__ Y,
                                                 const float* __restrict__ V,
                                                 float alpha, float beta) {
  __shared__ float ldsT[2][Cc * XT_STRIDE];

  const int lane = threadIdx.x & 31;
  const int wave = threadIdx.x >> 5;
  const int lm = lane & 15;
  const int lh = lane >> 4;
  const int b = blockIdx.y;
  const int r0 = blockIdx.x * 128 + wave * 16;

  const float* Wb = W + b * (Nn * Cc);

  // A-fragment row for this lane (clamp; OOB rows compute garbage, never stored)
  int arow = r0 + lm;
  if (arow > Nn - 1) arow = Nn - 1;
  const float* Srow = S + arow * Nn;

  v8f acc[4];
#pragma unroll
  for (int ct = 0; ct < 4; ++ct) acc[ct] = (v8f){};

  float4 pre[4];
  spmm_load_regs(Wb, 0, (Nn < 64 ? Nn : 64), pre);
  spmm_store_lds(ldsT[0], pre);

  int buf = 0;
  for (int k0 = 0; k0 < Nn; k0 += 64) {
    __syncthreads();  // current buffer fully written, visible to all waves

    const int kc = (Nn - k0 < 64) ? (Nn - k0) : 64;
    const bool more = (k0 + 64 < Nn);
    if (more) {  // prefetch next chunk into registers under the WMMA loop
      const int knext = k0 + 64;
      const int kcn = (Nn - knext < 64) ? (Nn - knext) : 64;
      spmm_load_regs(Wb, knext, kcn, pre);
    }

    const float* __restrict__ bufp = ldsT[buf];
    const int steps = kc >> 2;  // K advances by 4 per WMMA
    for (int s2 = 0; s2 < steps; ++s2) {
      const int kb = 4 * s2 + 2 * lh;              // this lane's first K index
      v2f a = *(const v2f*)(Srow + k0 + kb);       // 2 consecutive K, one b64
#pragma unroll
      for (int ct = 0; ct < 4; ++ct) {
        v2f bf = *(const v2f*)&bufp[(ct * 16 + lm) * XT_STRIDE + kb];
        acc[ct] = __builtin_amdgcn_wmma_f32_16x16x4_f32(
            false, a, false, bf, (short)0, acc[ct], false, false);
      }
    }

    if (more) spmm_store_lds(ldsT[buf ^ 1], pre);
    buf ^= 1;
  }

  // epilogue: Y = alpha*acc (+ beta*V), guarded on row < N
  const bool use_beta = (beta != 0.0f);
#pragma unroll
  for (int j = 0; j < 8; ++j) {
    const int row = r0 + j + 8 * lh;  // C/D layout: M = j + 8*(lane>=16)
    if (row < Nn) {
      const int base = b * (Nn * Cc) + row * Cc;
#pragma unroll
      for (int ct = 0; ct < 4; ++ct) {
        const int idx = base + ct * 16 + lm;
        float v = alpha * acc[ct][j];
        if (use_beta) v += beta * V[idx];
        Y[idx] = v;
      }
    }
  }
}

// ---------------------------------------------------------------------------
// Kernel 4: per-node combine.  Block = node n.
//   W_n[192,64] = sum_d emb[n,d]*pool[d,:,:,:]  (LDS, K-contiguous transposed)
//   out[b,n,o]  = sum_ki A[b,ki]*W_n[ki,o] + bias_n[o],  A = [x|Y1|Y2][b,n,:]
// 8 waves: wave = (mt,ct), mt in {0,1} batch-tile, ct in {0..3} out-col tile.
// ---------------------------------------------------------------------------
__global__ __launch_bounds__(256) void agcn_combine(
    const float* __restrict__ x, const float* __restrict__ y1,
    const float* __restrict__ y2, const float* __restrict__ emb,
    const float* __restrict__ pool, const float* __restrict__ bpool,
    float* __restrict__ out) {
  __shared__ float lwT[Cc * WT_STRIDE];  // transposed: lwT[o*194 + ki]
  __shared__ float lb[Cc];
  __shared__ float le[Ee];

  const int n = blockIdx.x;
  const int t = threadIdx.x;

  if (t < Ee) le[t] = emb[n * Ee + t];
  __syncthreads();

  for (int idx = t; idx < KI * Cc; idx += 256) {  // idx = ki*64 + o
    float a = 0.0f;
#pragma unroll
    for (int d = 0; d < Ee; ++d) a += le[d] * pool[d * (KI * Cc) + idx];
    const int ki = idx >> 6;
    const int o = idx & 63;
    lwT[o * WT_STRIDE + ki] = a;
  }
  if (t < Cc) {
    float a = 0.0f;
#pragma unroll
    for (int d = 0; d < Ee; ++d) a += le[d] * bpool[d * Cc + t];
    lb[t] = a;
  }
  __syncthreads();

  const int lane = t & 31;
  const int wave = t >> 5;
  const int mt = wave >> 2;
  const int ct = wave & 3;
  const int lm = lane & 15;
  const int lh = lane >> 4;

  const int brow = mt * 16 + lm;            // batch index for A fragment
  const int rowoff = brow * (Nn * Cc) + n * Cc;
  const int col = ct * 16 + lm;             // output feature for B/C fragments

  v8f acc = (v8f){};
  for (int s2 = 0; s2 < KI / 4; ++s2) {
    const int kk = 4 * s2 + 2 * lh;         // K index in [0,192)
    const int blk = kk >> 6;                // source select (wave-uniform)
    const int i = kk & 63;
    const float* p = (blk == 0) ? x : ((blk == 1) ? y1 : y2);
    v2f a = *(const v2f*)(p + rowoff + i);
    v2f bf = *(const v2f*)&lwT[col * WT_STRIDE + kk];  // single b64, K-contig
    acc = __builtin_amdgcn_wmma_f32_16x16x4_f32(
        false, a, false, bf, (short)0, acc, false, false);
  }

  const float bias = lb[col];
  const int outbase = n * Cc + col;
#pragma unroll
  for (int j = 0; j < 8; ++j) {
    const int bb = mt * 16 + j + 8 * lh;
    out[bb * (Nn * Cc) + outbase] = acc[j] + bias;
  }
}

// ---------------------------------------------------------------------------
extern "C" void kernel_launch(void* const* d_in, const int* in_sizes, int n_in,
                              void* d_out, int out_size, void* d_ws, size_t ws_size,
                              hipStream_t stream) {
  const float* x     = (const float*)d_in[0];  // [B,N,64]
  const float* emb   = (const float*)d_in[1];  // [N,16]
  const float* nv1   = (const float*)d_in[2];  // [N,16]
  const float* nv2   = (const float*)d_in[3];  // [16,N]
  const float* pool  = (const float*)d_in[4];  // [16,3,64,64]
  const float* bpool = (const float*)d_in[5];  // [16,64]
  float* out = (float*)d_out;                  // [B,N,64]

  float* ws = (float*)d_ws;
  float* S  = ws;                      // N*N   = 4,000,000 floats
  float* Y1 = S + (size_t)Nn * Nn;     // B*N*C = 4,096,000 floats
  float* Y2 = Y1 + (size_t)Bb * Nn * Cc;

  // 1) adjacency softmax
  agcn_adj<<<Nn, 256, 0, stream>>>(nv1, nv2, S);

  // 2) Y1 = S @ x           3) Y2 = 2*S@Y1 - x
  dim3 g((Nn + 127) / 128, Bb);
  agcn_spmm<<<g, 256, 0, stream>>>(S, x,  Y1, x, 1.0f,  0.0f);
  agcn_spmm<<<g, 256, 0, stream>>>(S, Y1, Y2, x, 2.0f, -1.0f);

  // 4) per-node weighted combine + bias
  agcn_combine<<<Nn, 256, 0, stream>>>(x, Y1, Y2, emb, pool, bpool, out);
}